// MLP_learner_30227979829652
// MI455X (gfx1250) — compile-verified
//
#include <hip/hip_runtime.h>
#include <hip/hip_bf16.h>

#define NN 8192
#define DD 128

typedef float v2f __attribute__((ext_vector_type(2)));
typedef float v8f __attribute__((ext_vector_type(8)));

// ---------------------------------------------------------------------------
// Kernel 1: emb = L2normalize( relu(X @ W1^T + b1) @ W2^T + b2 )
// One 128-thread workgroup per row. W1/W2 (64KB each) stay L2-resident.
// ---------------------------------------------------------------------------
__global__ __launch_bounds__(128) void emb_kernel(
    const float* __restrict__ X,
    const float* __restrict__ W1, const float* __restrict__ b1,
    const float* __restrict__ W2, const float* __restrict__ b2,
    float* __restrict__ emb) {
  __shared__ float sx[DD];
  __shared__ float sh[DD];
  __shared__ float red[4];

  const int row = blockIdx.x;
  const int j = threadIdx.x;

  sx[j] = X[(size_t)row * DD + j];
  __syncthreads();

  // h[j] = relu(dot(x, W1[j]) + b1[j])
  const float* __restrict__ w1 = W1 + (size_t)j * DD;
  float acc = b1[j];
#pragma unroll 8
  for (int k = 0; k < DD; ++k) acc = fmaf(sx[k], w1[k], acc);
  sh[j] = fmaxf(acc, 0.f);
  __syncthreads();

  // e[j] = dot(h, W2[j]) + b2[j]
  const float* __restrict__ w2 = W2 + (size_t)j * DD;
  float e = b2[j];
#pragma unroll 8
  for (int k = 0; k < DD; ++k) e = fmaf(sh[k], w2[k], e);

  // row L2 norm (4 waves of 32)
  float sq = e * e;
#pragma unroll
  for (int off = 16; off > 0; off >>= 1) sq += __shfl_down(sq, off, 32);
  if ((j & 31) == 0) red[j >> 5] = sq;
  __syncthreads();
  float tot = red[0] + red[1] + red[2] + red[3];
  float inv = rsqrtf(fmaxf(tot, 1e-24f));  // clip(norm,1e-12) equivalent
  emb[(size_t)row * DD + j] = e * inv;
}

// ---------------------------------------------------------------------------
// Kernel 2: sim = emb @ emb^T using V_WMMA_F32_16X16X4_F32.
// Workgroup = 256 threads (8 waves) -> 128x128 output tile.
// Wave w (mg = w&3, ng = w>>2) computes a 32x64 strip: 2x4 v8f accumulators.
// Per k-step: 2 A frags + 4 B frags feed 8 WMMAs -> 192 B LDS / WMMA.
// K staged through LDS in 32-wide panels, row pitch 36 floats (16B aligned,
// 16 distinct banks across lanes 0..15 on fragment reads).
// A fragment layout per ISA: lanes 0-15 hold K=0,1 ; lanes 16-31 hold K=2,3.
// B is loaded with the symmetric pattern (Gram matrix => consistent pairing).
// ---------------------------------------------------------------------------
__global__ __launch_bounds__(256) void sim_wmma_kernel(
    const float* __restrict__ emb, float* __restrict__ out) {
  __shared__ float sA[128 * 36];
  __shared__ float sB[128 * 36];

  const int tid  = threadIdx.x;
  const int wave = tid >> 5;
  const int lane = tid & 31;
  const int lo   = lane & 15;
  const int hi   = lane >> 4;
  const int mg   = wave & 3;   // row group: 32 rows each
  const int ng   = wave >> 2;  // col group: 64 cols each

  const int rBase = blockIdx.y * 128;
  const int cBase = blockIdx.x * 128;

  v8f acc[2][4];
#pragma unroll
  for (int mi = 0; mi < 2; ++mi)
#pragma unroll
    for (int t = 0; t < 4; ++t) acc[mi][t] = (v8f){};

  for (int k0 = 0; k0 < DD; k0 += 32) {
    __syncthreads();
    {  // stage panels: 128 rows x 32 K each (A from rBase, B from cBase)
      const int r = tid >> 1, half = tid & 1;
      const float4* srcA =
          (const float4*)(emb + (size_t)(rBase + r) * DD + k0 + half * 16);
      float4* dstA = (float4*)(sA + r * 36 + half * 16);
      dstA[0] = srcA[0]; dstA[1] = srcA[1]; dstA[2] = srcA[2]; dstA[3] = srcA[3];
      const float4* srcB =
          (const float4*)(emb + (size_t)(cBase + r) * DD + k0 + half * 16);
      float4* dstB = (float4*)(sB + r * 36 + half * 16);
      dstB[0] = srcB[0]; dstB[1] = srcB[1]; dstB[2] = srcB[2]; dstB[3] = srcB[3];
    }
    __syncthreads();

#pragma unroll
    for (int kk = 0; kk < 32; kk += 4) {
      v2f a[2];
#pragma unroll
      for (int mi = 0; mi < 2; ++mi) {
        const int arow = mg * 32 + mi * 16 + lo;
        a[mi].x = sA[arow * 36 + kk + 2 * hi];
        a[mi].y = sA[arow * 36 + kk + 2 * hi + 1];
      }
#pragma unroll
      for (int t = 0; t < 4; ++t) {
        v2f b;
        const int brow = ng * 64 + t * 16 + lo;
        b.x = sB[brow * 36 + kk + 2 * hi];
        b.y = sB[brow * 36 + kk + 2 * hi + 1];
        // 8 args: (neg_a, A, neg_b, B, c_mod, C, reuse_a, reuse_b)
        acc[0][t] = __builtin_amdgcn_wmma_f32_16x16x4_f32(
            false, a[0], false, b, (short)0, acc[0][t], false, false);
        acc[1][t] = __builtin_amdgcn_wmma_f32_16x16x4_f32(
            false, a[1], false, b, (short)0, acc[1][t], false, false);
      }
    }
  }

  // C/D layout: VGPR r -> M = r + 8*hi, N = lo
#pragma unroll
  for (int mi = 0; mi < 2; ++mi) {
    const int mBase = rBase + mg * 32 + mi * 16 + 8 * hi;
#pragma unroll
    for (int t = 0; t < 4; ++t) {
#pragma unroll
      for (int r = 0; r < 8; ++r) {
        out[(size_t)(mBase + r) * NN + cBase + ng * 64 + t * 16 + lo] =
            acc[mi][t][r];
      }
    }
  }
}

// ---------------------------------------------------------------------------
// Kernel 3: per-row top-kp1 mask + relu, in place on out.
// One 256-thread workgroup per row; row cached in LDS as order-preserving
// uint keys; 32-round radix bisection finds the kp1-th largest value exactly.
// ---------------------------------------------------------------------------
__global__ __launch_bounds__(256) void topk_relu_kernel(
    float* __restrict__ out, const int* __restrict__ kp1p) {
  __shared__ unsigned urow[NN];
  __shared__ int cnt;

  const int r = blockIdx.x;
  const int tid = threadIdx.x;
  float* __restrict__ orow = out + (size_t)r * NN;
  const int kp1 = *kp1p;

  // load + map float -> order-preserving uint
  for (int i = tid; i < NN / 4; i += 256) {
    float4 v = ((const float4*)orow)[i];
    unsigned b;
    b = __float_as_uint(v.x); urow[4*i+0] = (b & 0x80000000u) ? ~b : (b | 0x80000000u);
    b = __float_as_uint(v.y); urow[4*i+1] = (b & 0x80000000u) ? ~b : (b | 0x80000000u);
    b = __float_as_uint(v.z); urow[4*i+2] = (b & 0x80000000u) ? ~b : (b | 0x80000000u);
    b = __float_as_uint(v.w); urow[4*i+3] = (b & 0x80000000u) ? ~b : (b | 0x80000000u);
  }
  __syncthreads();

  // find max threshold t with count(u >= t) >= kp1  (== key of kp1-th largest)
  unsigned result = 0u;
  for (int bit = 31; bit >= 0; --bit) {
    const unsigned cand = result | (1u << bit);
    if (tid == 0) cnt = 0;
    __syncthreads();
    int c = 0;
    for (int i = tid; i < NN; i += 256) c += (urow[i] >= cand) ? 1 : 0;
    atomicAdd(&cnt, c);
    __syncthreads();
    if (cnt >= kp1) result = cand;
    __syncthreads();
  }

  // keep top-kp1, apply relu (kept value > 0 <=> key > 0x80000000)
  for (int i = tid; i < NN; i += 256) {
    const unsigned u = urow[i];
    float o = 0.f;
    if (u >= result && u > 0x80000000u) o = __uint_as_float(u ^ 0x80000000u);
    orow[i] = o;
  }
}

// ---------------------------------------------------------------------------
extern "C" void kernel_launch(void* const* d_in, const int* in_sizes, int n_in,
                              void* d_out, int out_size, void* d_ws, size_t ws_size,
                              hipStream_t stream) {
  const float* X  = (const float*)d_in[0];
  const float* W1 = (const float*)d_in[1];
  const float* b1 = (const float*)d_in[2];
  const float* W2 = (const float*)d_in[3];
  const float* b2 = (const float*)d_in[4];
  const int* kp1  = (const int*)d_in[5];
  float* out = (float*)d_out;
  float* emb = (float*)d_ws;  // 8192*128*4 = 4 MB scratch

  emb_kernel<<<NN, 128, 0, stream>>>(X, W1, b1, W2, b2, emb);

  dim3 g2(NN / 128, NN / 128);  // 64 x 64 workgroups, 128x128 tiles
  sim_wmma_kernel<<<g2, 256, 0, stream>>>(emb, out);

  topk_relu_kernel<<<NN, 256, 0, stream>>>(out, kp1);
}